// LAPLoss_14027363188886
// MI455X (gfx1250) — compile-verified
//
#include <hip/hip_runtime.h>

typedef __attribute__((ext_vector_type(2))) float v2f;
typedef __attribute__((ext_vector_type(8))) float v8f;
typedef __attribute__((ext_vector_type(2))) int   v2i;

// Wave32 reduction using V_WMMA_F32_16X16X4_F32.
// A: lane L<16 -> A[L,0]=p, A[L,1]=0 ; lane L>=16 -> A[L-16,2]=p, A[L-16,3]=0
// B: all ones (layout-independent).  D[m,n] = p_m + p_{m+16} for every n.
// Sum of a lane's 8 D VGPRs = sum of 8 consecutive rows; add the value from
// the other 16-lane half (xor 16) -> full 32-lane total in every lane.
__device__ __forceinline__ float wave_reduce_f32(float p) {
    v2f a; a[0] = p;    a[1] = 0.0f;
    v2f b; b[0] = 1.0f; b[1] = 1.0f;
    v8f c = {};
    v8f d = __builtin_amdgcn_wmma_f32_16x16x4_f32(
        /*neg_a=*/false, a, /*neg_b=*/false, b,
        /*c_mod=*/(short)0, c, /*reuse_a=*/false, /*reuse_b=*/false);
    float t = ((d[0] + d[1]) + (d[2] + d[3])) + ((d[4] + d[5]) + (d[6] + d[7]));
    t += __shfl_xor(t, 16, 32);
    return t;
}

__global__ __launch_bounds__(256) void lap_reg_kernel(
    const float* __restrict__ x1, const float* __restrict__ x2,
    const int*  __restrict__ lap, int n, float scale,
    float* __restrict__ out)
{
    float p = 0.0f;
    const int tid    = blockIdx.x * blockDim.x + threadIdx.x;
    const int stride = gridDim.x * blockDim.x;

    for (int i = tid; i < n; i += stride) {
        // lap row: 10 ints, byte offset 40*i -> always 8B aligned: five b64 loads.
        const v2i* row = (const v2i*)(lap + (long long)i * 10);
        v2i i01 = __builtin_nontemporal_load(row + 0);
        v2i i23 = __builtin_nontemporal_load(row + 1);
        v2i i45 = __builtin_nontemporal_load(row + 2);
        v2i i67 = __builtin_nontemporal_load(row + 3);
        v2i pc  = __builtin_nontemporal_load(row + 4);   // [pad, count]

        // self coords: e = x1 - x2
        const long long b = (long long)i * 3;
        float ex = x1[b + 0] - x2[b + 0];
        float ey = x1[b + 1] - x2[b + 1];
        float ez = x1[b + 2] - x2[b + 2];

        int idxs[8] = { i01[0], i01[1], i23[0], i23[1],
                        i45[0], i45[1], i67[0], i67[1] };

        float sx = 0.0f, sy = 0.0f, sz = 0.0f;
#pragma unroll
        for (int j = 0; j < 8; ++j) {
            int v = idxs[j];
            if (v >= 0) {                       // -1 marks invalid neighbor
                const long long o = (long long)v * 3;
                sx += x1[o + 0] - x2[o + 0];
                sy += x1[o + 1] - x2[o + 1];
                sz += x1[o + 2] - x2[o + 2];
            }
        }

        const float inv = 1.0f / (float)pc[1];  // count >= 1 always
        const float dx = ex - sx * inv;
        const float dy = ey - sy * inv;
        const float dz = ez - sz * inv;
        p += dx * dx + dy * dy + dz * dz;
    }

    p *= scale;

    // ---- wave32 reduce (WMMA), EXEC is all-ones here (no divergence) ----
    const int lane = threadIdx.x & 31;
    const int wave = threadIdx.x >> 5;
    float t = wave_reduce_f32(p);

    __shared__ float wsum[8];                   // 256 threads = 8 waves
    if (lane == 0) wsum[wave] = t;
    __syncthreads();

    if (wave == 0) {
        float s = (lane < 8) ? wsum[lane] : 0.0f;
        s += __shfl_xor(s, 4, 32);
        s += __shfl_xor(s, 2, 32);
        s += __shfl_xor(s, 1, 32);
        if (lane == 0) atomicAdd(out, s);
    }
}

extern "C" void kernel_launch(void* const* d_in, const int* in_sizes, int n_in,
                              void* d_out, int out_size, void* d_ws, size_t ws_size,
                              hipStream_t stream) {
    const float* coarse_in   = (const float*)d_in[0];
    const float* coarse_pred = (const float*)d_in[1];
    const float* fine_in     = (const float*)d_in[2];
    const float* fine_pred   = (const float*)d_in[3];
    const int*   lap_coarse  = (const int*)d_in[4];
    const int*   lap_fine    = (const int*)d_in[5];
    float* out = (float*)d_out;

    const int n_c = in_sizes[0] / 3;
    const int n_f = in_sizes[2] / 3;

    // Output accumulator must start at zero each call (capture-safe memset node).
    hipMemsetAsync(d_out, 0, (size_t)out_size * sizeof(float), stream);

    const int block = 256;
    int grid_c = (n_c + block - 1) / block; if (grid_c > 2048) grid_c = 2048;
    int grid_f = (n_f + block - 1) / block; if (grid_f > 4096) grid_f = 4096;
    if (grid_c < 1) grid_c = 1;
    if (grid_f < 1) grid_f = 1;

    lap_reg_kernel<<<grid_c, block, 0, stream>>>(
        coarse_in, coarse_pred, lap_coarse, n_c, 0.5f / (float)n_c, out);
    lap_reg_kernel<<<grid_f, block, 0, stream>>>(
        fine_in, fine_pred, lap_fine, n_f, 0.5f / (float)n_f, out);
}